// Softsplat_35021163331685
// MI455X (gfx1250) — compile-verified
//
#include <hip/hip_runtime.h>
#include <math.h>

// Softsplat (forward warp with exp(z) metric) for MI455X / gfx1250.
// N=4, C=64, H=256, W=448, all float32.
//
// Pass 1: zero out[] (64ch accumulator) and norm[] (metric accumulator, in d_ws)
// Pass 2: scatter — one thread per source pixel, 4 bilinear corners x 65 channels
//         of non-returning relaxed agent-scope global_atomic_add_f32.
//         Accumulators (119 MB) fit in the 192 MB L2, so atomics resolve on-chip.
// Pass 3: normalize — out /= (norm == 0 ? 1 : norm), float4-vectorized.

namespace {
constexpr int N = 4;
constexpr int C = 64;
constexpr int H = 256;
constexpr int W = 448;
constexpr int HW = H * W;                 // 114688
constexpr int NPIX = N * HW;              // 458752
constexpr int OUT_ELEMS = N * C * HW;     // 29360128
}

// Non-returning native FP32 global atomic add (relaxed, device scope).
// Discarding the result lets the backend drop TH=RET -> pure store-side atomic.
__device__ __forceinline__ void atomic_add_f32(float* p, float v) {
  (void)__hip_atomic_fetch_add(p, v, __ATOMIC_RELAXED, __HIP_MEMORY_SCOPE_AGENT);
}

__global__ __launch_bounds__(256) void zero_f4(float4* __restrict__ p, int n4) {
  int i = blockIdx.x * blockDim.x + threadIdx.x;
  if (i < n4) p[i] = float4{0.f, 0.f, 0.f, 0.f};
}

__global__ __launch_bounds__(256) void softsplat_scatter(
    const float* __restrict__ img, const float* __restrict__ flow,
    const float* __restrict__ z, float* __restrict__ out,
    float* __restrict__ norm) {
  const int t = blockIdx.x * blockDim.x + threadIdx.x;
  if (t >= NPIX) return;
  const int n  = t / HW;
  const int hw = t - n * HW;
  const int y  = hw / W;
  const int x  = hw - y * W;

  // Target position (matches reference: finite-mask, floor, per-corner masks).
  const float fx = (float)x + flow[(n * 2 + 0) * HW + hw];
  const float fy = (float)y + flow[(n * 2 + 1) * HW + hw];
  const float inf = __builtin_inff();
  const bool finite = (fabsf(fx) < inf) && (fabsf(fy) < inf);
  const float sx = finite ? fx : 0.f;
  const float sy = finite ? fy : 0.f;

  const float x0f = floorf(sx), y0f = floorf(sy);
  const float x1f = x0f + 1.f,  y1f = y0f + 1.f;
  const int x0 = (int)x0f, y0 = (int)y0f;
  const int x1 = x0 + 1,   y1 = y0 + 1;

  const float ax = sx - x0f, ay = sy - y0f;   // fractional parts
  const float bx = x1f - sx, by = y1f - sy;

  const bool inx0 = (x0 >= 0) & (x0 < W);
  const bool inx1 = (x1 >= 0) & (x1 < W);
  const bool iny0 = (y0 >= 0) & (y0 < H);
  const bool iny1 = (y1 >= 0) & (y1 < H);

  const float e = expf(z[n * HW + hw]);       // importance metric

  // Fold metric into the (masked) bilinear weights once.
  const float we00 = (finite & inx0 & iny0) ? (bx * by) * e : 0.f;
  const float we10 = (finite & inx1 & iny0) ? (ax * by) * e : 0.f;
  const float we01 = (finite & inx0 & iny1) ? (bx * ay) * e : 0.f;
  const float we11 = (finite & inx1 & iny1) ? (ax * ay) * e : 0.f;

  const bool a00 = (we00 != 0.f);
  const bool a10 = (we10 != 0.f);
  const bool a01 = (we01 != 0.f);
  const bool a11 = (we11 != 0.f);

  const int x0c = min(max(x0, 0), W - 1);
  const int x1c = min(max(x1, 0), W - 1);
  const int y0c = min(max(y0, 0), H - 1);
  const int y1c = min(max(y1, 0), H - 1);
  const int p00 = y0c * W + x0c;
  const int p10 = y0c * W + x1c;
  const int p01 = y1c * W + x0c;
  const int p11 = y1c * W + x1c;

  // Normalizer channel (skip zero-weight adds: adding +0.0 is a no-op).
  float* nb = norm + n * HW;
  if (a00) atomic_add_f32(nb + p00, we00);
  if (a10) atomic_add_f32(nb + p10, we10);
  if (a01) atomic_add_f32(nb + p01, we01);
  if (a11) atomic_add_f32(nb + p11, we11);

  // 64 image channels: coalesced strided loads + coalesced L2 atomics.
  const float* ip = img + (n * C) * HW + hw;
  float* ob = out + (n * C) * HW;
#pragma unroll 4
  for (int c = 0; c < C; ++c) {
    if (c + 4 < C) __builtin_prefetch(ip + (c + 4) * HW, 0, 0);  // global_prefetch_b8
    const float v = ip[c * HW];
    float* oc = ob + c * HW;
    if (a00) atomic_add_f32(oc + p00, v * we00);
    if (a10) atomic_add_f32(oc + p10, v * we10);
    if (a01) atomic_add_f32(oc + p01, v * we01);
    if (a11) atomic_add_f32(oc + p11, v * we11);
  }
}

__global__ __launch_bounds__(256) void softsplat_normalize(
    float4* __restrict__ out, const float* __restrict__ norm) {
  const int i = blockIdx.x * blockDim.x + threadIdx.x;  // float4 index
  constexpr int TOTAL4 = OUT_ELEMS / 4;
  if (i >= TOTAL4) return;
  const int e0  = i * 4;
  const int n   = e0 / (C * HW);
  const int rem = e0 - n * (C * HW);
  const int hw  = rem % HW;                 // multiple of 4 (HW % 4 == 0)

  float4 o = out[i];
  float4 w = *(const float4*)(norm + n * HW + hw);
  w.x = (w.x == 0.f) ? 1.f : w.x;
  w.y = (w.y == 0.f) ? 1.f : w.y;
  w.z = (w.z == 0.f) ? 1.f : w.z;
  w.w = (w.w == 0.f) ? 1.f : w.w;
  o.x /= w.x; o.y /= w.y; o.z /= w.z; o.w /= w.w;
  out[i] = o;
}

extern "C" void kernel_launch(void* const* d_in, const int* in_sizes, int n_in,
                              void* d_out, int out_size, void* d_ws, size_t ws_size,
                              hipStream_t stream) {
  (void)in_sizes; (void)n_in; (void)out_size; (void)ws_size;
  const float* img  = (const float*)d_in[0];
  const float* flow = (const float*)d_in[1];
  const float* z    = (const float*)d_in[2];
  float* out  = (float*)d_out;
  float* norm = (float*)d_ws;   // N*HW floats = 1.84 MB of scratch

  // Zero accumulators every call (harness does not re-poison between replays).
  {
    const int n4 = OUT_ELEMS / 4;
    zero_f4<<<(n4 + 255) / 256, 256, 0, stream>>>((float4*)out, n4);
  }
  {
    const int n4 = NPIX / 4;
    zero_f4<<<(n4 + 255) / 256, 256, 0, stream>>>((float4*)norm, n4);
  }

  softsplat_scatter<<<(NPIX + 255) / 256, 256, 0, stream>>>(img, flow, z, out, norm);

  {
    const int n4 = OUT_ELEMS / 4;
    softsplat_normalize<<<(n4 + 255) / 256, 256, 0, stream>>>((float4*)out, norm);
  }
}